// PANConvEdgeClassifier_86938728005825
// MI455X (gfx1250) — compile-verified
//
#include <hip/hip_runtime.h>

typedef float v2f __attribute__((ext_vector_type(2)));
typedef float v8f __attribute__((ext_vector_type(8)));

#define HID 64
#define HASH_BITS 22
#define HASH_N (1u << HASH_BITS)
#define HASH_MASK (HASH_N - 1u)
#define EMPTY_KEY 0xFFFFFFFFFFFFFFFFULL

// ---------------------------------------------------------------- utilities

__global__ void fill_f32_kernel(float* p, float v, int n) {
    int i = blockIdx.x * blockDim.x + threadIdx.x;
    if (i < n) p[i] = v;
}

__global__ void hash_init_kernel(unsigned long long* tab) {
    int i = blockIdx.x * blockDim.x + threadIdx.x;
    if (i < (int)HASH_N) tab[i] = EMPTY_KEY;
}

__device__ __forceinline__ unsigned mix64(unsigned long long x) {
    x ^= x >> 33; x *= 0xff51afd7ed558ccdULL;
    x ^= x >> 33; x *= 0xc4ceb9fe1a85ec53ULL;
    x ^= x >> 33;
    return (unsigned)x;
}

// distinct (dst,src) pairs of A^T; diagonal handled by rowcount init = 1.0
__global__ void edge_dedup_kernel(const int* __restrict__ src,
                                  const int* __restrict__ dst, int E,
                                  unsigned long long* __restrict__ tab,
                                  float* __restrict__ rowcount) {
    int e = blockIdx.x * blockDim.x + threadIdx.x;
    if (e >= E) return;
    int s = src[e], d = dst[e];
    if (s == d) return;  // duplicates the identity diagonal
    unsigned long long key = ((unsigned long long)(unsigned)d << 32) | (unsigned)s;
    unsigned h = mix64(key) & HASH_MASK;
    for (;;) {
        unsigned long long prev = atomicCAS(&tab[h], EMPTY_KEY, key);
        if (prev == EMPTY_KEY) { atomicAdd(&rowcount[d], 1.0f); return; }
        if (prev == key) return;  // duplicate edge
        h = (h + 1u) & HASH_MASK;
    }
}

__global__ void invert_kernel(float* p, int n) {
    int i = blockIdx.x * blockDim.x + threadIdx.x;
    if (i < n) p[i] = 1.0f / p[i];   // rowcount >= 1 always
}

// ---------------------------------------------------------------- WMMA GEMM
// Y[nrows,64] = X[nrows,K] @ W[K,64] + bias.  One wave -> 16x64 stripe.
// A layout (f32 16x4): lane m=lane&15, half=lane>>4; v0=K(2*half), v1=K(2*half+1)
// B layout (f32 4x16): lane n=lane&15; v0=W[2*half][n], v1=W[2*half+1][n]
// C/D (f32 16x16): VGPR v, lane -> row v+8*half, col lane&15
//
// W is staged in LDS in K-pair-interleaved form:
//   sW[(k>>1)*128 + 2*n + (k&1)] = W[k][n]
// so each lane's (W[ka][n], W[ka+1][n]) B operand is one aligned ds_load_b64.
template <int K>
__global__ __launch_bounds__(256) void gemm_wmma_kernel(
    const float* __restrict__ X, const float* __restrict__ W,
    const float* __restrict__ bias, float* __restrict__ Y, int ntiles) {
    __shared__ float sW[K * 64];
    for (int i = threadIdx.x; i < K * 64; i += 256) {
        int k = i >> 6, n = i & 63;
        sW[(k >> 1) * 128 + 2 * n + (k & 1)] = W[i];
    }
    __syncthreads();

    const int lane = threadIdx.x & 31;
    const int tile = blockIdx.x * 8 + (threadIdx.x >> 5);
    if (tile >= ntiles) return;  // wave-uniform: EXEC stays all-ones for WMMA

    const int m = lane & 15;
    const int half = lane >> 4;
    const long row0 = (long)tile * 16;
    const float* xrow = X + (row0 + m) * K;

    v8f acc[4] = {};
    for (int k0 = 0; k0 < K; k0 += 4) {
        const int ka = k0 + 2 * half;           // even
        v2f a = *(const v2f*)(xrow + ka);       // 8B aligned
        const float* wrow = sW + (ka >> 1) * 128 + 2 * m;
#pragma unroll
        for (int t = 0; t < 4; ++t) {
            v2f b = *(const v2f*)(wrow + 32 * t);  // ds_load_b64
            acc[t] = __builtin_amdgcn_wmma_f32_16x16x4_f32(
                false, a, false, b, (short)0, acc[t], false, false);
        }
    }
#pragma unroll
    for (int t = 0; t < 4; ++t) {
        const int n = t * 16 + m;
        const float bv = bias[n];
#pragma unroll
        for (int v = 0; v < 8; ++v)
            Y[(row0 + v + 8 * half) * HID + n] = acc[t][v] + bv;
    }
}

// ---------------------------------------------------------------- edge scatter
// agg[dst] += h[src]; thread covers (edge, feature-pair) -> coalesced 256B/wave
__global__ __launch_bounds__(256) void scatter_kernel(
    const float* __restrict__ h, const int* __restrict__ src,
    const int* __restrict__ dst, float* __restrict__ agg, int E) {
    int tid = blockIdx.x * blockDim.x + threadIdx.x;
    if (tid >= E * 32) return;
    int e = tid >> 5;
    int j = (tid & 31) * 2;
    int s = src[e], d = dst[e];
    v2f hv = *(const v2f*)(h + (long)s * HID + j);
    atomicAdd(agg + (long)d * HID + j, hv.x);
    atomicAdd(agg + (long)d * HID + j + 1, hv.y);
}

// out = norm * (p0*h + p0*p1*agg), optional ReLU (fused elementwise)
__global__ void combine_kernel(const float* __restrict__ h,
                               const float* __restrict__ agg,
                               const float* __restrict__ norm,
                               const float* __restrict__ wpan,
                               float* __restrict__ out, int total, int relu) {
    int i = blockIdx.x * blockDim.x + threadIdx.x;
    if (i >= total) return;
    float p0 = wpan[0], p01 = wpan[0] * wpan[1];
    float v = norm[i >> 6] * (p0 * h[i] + p01 * agg[i]);
    if (relu) v = fmaxf(v, 0.0f);
    out[i] = v;
}

// ---------------------------------------------------------------- classifier
// out[e,:] = concat(h[row], h[col]) @ wc + bc ; one wave per edge
__global__ __launch_bounds__(256) void classify_kernel(
    const float* __restrict__ h, const int* __restrict__ row,
    const int* __restrict__ col, const float* __restrict__ wc,
    const float* __restrict__ bc, float* __restrict__ out, int E) {
    const int lane = threadIdx.x & 31;
    const int e = blockIdx.x * 8 + (threadIdx.x >> 5);
    if (e >= E) return;
    int r = row[e], c = col[e];
    v2f a = *(const v2f*)(h + (long)r * HID + 2 * lane);
    v2f b = *(const v2f*)(h + (long)c * HID + 2 * lane);
    float4 w0 = *(const float4*)(wc + 4 * lane);        // wc rows 2l, 2l+1
    float4 w1 = *(const float4*)(wc + 128 + 4 * lane);  // wc rows 64+2l, 65+2l
    float p0 = a.x * w0.x + a.y * w0.z + b.x * w1.x + b.y * w1.z;
    float p1 = a.x * w0.y + a.y * w0.w + b.x * w1.y + b.y * w1.w;
#pragma unroll
    for (int off = 16; off; off >>= 1) {
        p0 += __shfl_xor(p0, off, 32);
        p1 += __shfl_xor(p1, off, 32);
    }
    if (lane == 0) {
        out[(long)e * 2 + 0] = p0 + bc[0];
        out[(long)e * 2 + 1] = p1 + bc[1];
    }
}

// ---------------------------------------------------------------- launch

extern "C" void kernel_launch(void* const* d_in, const int* in_sizes, int n_in,
                              void* d_out, int out_size, void* d_ws, size_t ws_size,
                              hipStream_t stream) {
    const float* x    = (const float*)d_in[0];
    const int*   eidx = (const int*)d_in[1];
    const float* w1   = (const float*)d_in[2];
    const float* b1   = (const float*)d_in[3];
    const float* p1   = (const float*)d_in[4];
    const float* w2   = (const float*)d_in[5];
    const float* b2   = (const float*)d_in[6];
    const float* p2   = (const float*)d_in[7];
    const float* wc   = (const float*)d_in[8];
    const float* bc   = (const float*)d_in[9];
    float* out = (float*)d_out;

    const int N = in_sizes[0] / 128;
    const int E = in_sizes[1] / 2;
    const int* src = eidx;       // edge_index[0]
    const int* dst = eidx + E;   // edge_index[1]

    // workspace layout (hash table aliases bufA/bufB; used before they exist)
    char* base = (char*)d_ws;
    size_t o = 0;
    float* norm = (float*)(base + o);
    o += ((size_t)N * 4 + 255) & ~(size_t)255;
    const size_t fbytes = ((size_t)N * HID * 4 + 255) & ~(size_t)255;
    float* bufA = (float*)(base + o); o += fbytes;   // h
    float* bufB = (float*)(base + o); o += fbytes;   // agg
    float* bufC = (float*)(base + o);                // layer output
    unsigned long long* hash = (unsigned long long*)bufA;  // 32MB < 2*fbytes

    const int NH = N * HID;
    const int ntiles = (N + 15) / 16;

    // norm = 1 / distinct-rowcount(I + A^T)
    hash_init_kernel<<<(HASH_N + 255) / 256, 256, 0, stream>>>(hash);
    fill_f32_kernel<<<(N + 255) / 256, 256, 0, stream>>>(norm, 1.0f, N);
    edge_dedup_kernel<<<(E + 255) / 256, 256, 0, stream>>>(src, dst, E, hash, norm);
    invert_kernel<<<(N + 255) / 256, 256, 0, stream>>>(norm, N);

    // layer 1: h1 = x@w1+b1 ; agg = A^T h1 ; out1 = relu(norm*(p0 h1 + p0 p1 agg))
    gemm_wmma_kernel<128><<<(ntiles + 7) / 8, 256, 0, stream>>>(x, w1, b1, bufA, ntiles);
    fill_f32_kernel<<<(NH + 255) / 256, 256, 0, stream>>>(bufB, 0.0f, NH);
    scatter_kernel<<<(E * 32 + 255) / 256, 256, 0, stream>>>(bufA, src, dst, bufB, E);
    combine_kernel<<<(NH + 255) / 256, 256, 0, stream>>>(bufA, bufB, norm, p1, bufC, NH, 1);

    // layer 2
    gemm_wmma_kernel<64><<<(ntiles + 7) / 8, 256, 0, stream>>>(bufC, w2, b2, bufA, ntiles);
    fill_f32_kernel<<<(NH + 255) / 256, 256, 0, stream>>>(bufB, 0.0f, NH);
    scatter_kernel<<<(E * 32 + 255) / 256, 256, 0, stream>>>(bufA, src, dst, bufB, E);
    combine_kernel<<<(NH + 255) / 256, 256, 0, stream>>>(bufA, bufB, norm, p2, bufC, NH, 0);

    // edge classifier
    classify_kernel<<<(E + 7) / 8, 256, 0, stream>>>(bufC, src, dst, wc, bc, out, E);
}